// CrossModalAttentionFusion_21260088115919
// MI455X (gfx1250) — compile-verified
//
#include <hip/hip_runtime.h>

// CrossModalAttentionFusion for MI455X (gfx1250, wave32).
// GEMMs via V_WMMA_F32_16X16X4_F32 (fp32-exact), attention/LN per-lane.
// v4: sched_barriers between token GEMMs to keep live VGPRs < 256
//     (kills s_set_vgpr_msb churn, restores 4 waves/SIMD occupancy).

typedef float v2f __attribute__((ext_vector_type(2)));
typedef float v8f __attribute__((ext_vector_type(8)));

#define NEGV (-1.0e9f)
#define LN_EPS 1e-5f

#define WAVES_PER_BLOCK 2
#define GROUPS_PER_WAVE 4
#define SAMPLES_PER_WAVE 16
#define SAMPLES_PER_BLOCK (WAVES_PER_BLOCK * GROUPS_PER_WAVE * SAMPLES_PER_WAVE) // 128
// per-wave LDS (floats): qkv[16][3][96] = 4608, ctx[16][3][32] = 1536
#define QKV_F 4608
#define CTX_F 1536
#define WAVE_LDS (QKV_F + CTX_F)

// Compile-time-foldable feature pointer select (t must be an unrolled constant)
#define FEAT(t) ((t) == 0 ? metric : ((t) == 1 ? logm : trace))

__global__ __launch_bounds__(64, 4) void cmaf_kernel(
    const float* __restrict__ metric, const float* __restrict__ logm,
    const float* __restrict__ trace, const int*   __restrict__ masks,
    const float* __restrict__ in_w,   const float* __restrict__ in_b,
    const float* __restrict__ out_w,  const float* __restrict__ out_b,
    const float* __restrict__ mweights, const float* __restrict__ gamma,
    const float* __restrict__ beta,   float* __restrict__ out)
{
    __shared__ float lds[WAVES_PER_BLOCK * WAVE_LDS];

    const int tid   = threadIdx.x;
    const int wave  = tid >> 5;
    const int lane  = tid & 31;
    const int lhalf = lane >> 4;   // 0: K/M low half, 1: high half (per WMMA layout)
    const int l16   = lane & 15;

    float* qkvs = &lds[wave * WAVE_LDS];   // qkv region; later reused for 'attended'
    float* ctxs = qkvs + QKV_F;            // ctx region

    const int blockStart = blockIdx.x * SAMPLES_PER_BLOCK;
    const float inv_sqrt_dh = 0.35355339059327373f;  // 1/sqrt(8)

    // ---- Preload all B-matrix weight fragments into registers (loop-invariant).
    // B layout (f32 16x16x4): lane L -> col N=L&15, K = 4c + 2*(L>>4) + {0,1}.
    v2f wb_in[6][8];    // in_proj_w fragments: 6 column tiles x 8 K-chunks
    v2f wb_out[2][8];   // out_proj_w fragments
    float bias_in[6], bias_out[2];
    #pragma unroll
    for (int ct = 0; ct < 6; ++ct) {
        const float* wrow = in_w + (size_t)(16 * ct + l16) * 32 + 2 * lhalf;
        #pragma unroll
        for (int c = 0; c < 8; ++c)
            wb_in[ct][c] = *(const v2f*)(wrow + 4 * c);
        bias_in[ct] = in_b[16 * ct + l16];
    }
    #pragma unroll
    for (int ct = 0; ct < 2; ++ct) {
        const float* wrow = out_w + (size_t)(16 * ct + l16) * 32 + 2 * lhalf;
        #pragma unroll
        for (int c = 0; c < 8; ++c)
            wb_out[ct][c] = *(const v2f*)(wrow + 4 * c);
        bias_out[ct] = out_b[16 * ct + l16];
    }
    __builtin_amdgcn_sched_barrier(0);

    for (int g = 0; g < GROUPS_PER_WAVE; ++g) {
        const int sbase = blockStart + (g * WAVES_PER_BLOCK + wave) * SAMPLES_PER_WAVE;

        // ============= Stage 1: qkv = feat @ in_proj_w^T + in_proj_b =========
        // A layout (f32 16x16x4): lane L -> row M=L&15, K = 4c + 2*(L>>4) + {0,1}.
        #pragma unroll
        for (int t = 0; t < 3; ++t) {
            v2f a[8];
            const float* frow = FEAT(t) + (size_t)(sbase + l16) * 32 + 2 * lhalf;
            #pragma unroll
            for (int c = 0; c < 8; ++c)
                a[c] = *(const v2f*)(frow + 4 * c);

            #pragma unroll
            for (int ct = 0; ct < 6; ++ct) {      // 6 column tiles cover 96 outputs
                v8f acc = {};
                #pragma unroll
                for (int c = 0; c < 8; ++c)        // K = 32 in 8 chunks of 4
                    acc = __builtin_amdgcn_wmma_f32_16x16x4_f32(
                            false, a[c], false, wb_in[ct][c], (short)0, acc, false, false);
                // D layout: lane L, reg r -> [M = 8*(L>>4)+r][N = L&15]
                #pragma unroll
                for (int r = 0; r < 8; ++r) {
                    const int m = 8 * lhalf + r;
                    qkvs[(m * 3 + t) * 96 + 16 * ct + l16] = acc[r] + bias_in[ct];
                }
            }
            // Serialize token pipelines: keeps one token's A-frags + 6 accs live
            // at a time so total pressure stays under the 256-VGPR window.
            __builtin_amdgcn_sched_barrier(0);
        }
        __syncthreads();

        // ============= Stage 2: masked multi-head attention (per-lane) =======
        {
            const int s  = l16;                 // sample within wave group
            const int sg = sbase + s;
            const int mk0 = masks[(size_t)sg * 3 + 0] != 0;
            const int mk1 = masks[(size_t)sg * 3 + 1] != 0;
            const int mk2 = masks[(size_t)sg * 3 + 2] != 0;
            const int mk[3] = { mk0, mk1, mk2 };

            #pragma unroll
            for (int hh = 0; hh < 2; ++hh) {    // each lane-half handles 2 heads
                const int h = 2 * lhalf + hh;
                float q[3][8], k[3][8], v[3][8];
                #pragma unroll
                for (int tt = 0; tt < 3; ++tt) {
                    const float* bp = &qkvs[(s * 3 + tt) * 96 + 8 * h];
                    #pragma unroll
                    for (int j = 0; j < 8; ++j) {
                        q[tt][j] = bp[j];
                        k[tt][j] = bp[32 + j];
                        v[tt][j] = bp[64 + j];
                    }
                }
                #pragma unroll
                for (int qt = 0; qt < 3; ++qt) {
                    float sc[3];
                    #pragma unroll
                    for (int kt = 0; kt < 3; ++kt) {
                        float d = 0.f;
                        #pragma unroll
                        for (int j = 0; j < 8; ++j) d += q[qt][j] * k[kt][j];
                        sc[kt] = mk[kt] ? d * inv_sqrt_dh : NEGV;
                    }
                    float rmax = fmaxf(sc[0], fmaxf(sc[1], sc[2]));
                    float e0 = __expf(sc[0] - rmax);
                    float e1 = __expf(sc[1] - rmax);
                    float e2 = __expf(sc[2] - rmax);
                    float rinv = 1.0f / (e0 + e1 + e2);
                    e0 *= rinv; e1 *= rinv; e2 *= rinv;
                    float* cp = &ctxs[(s * 3 + qt) * 32 + 8 * h];
                    #pragma unroll
                    for (int j = 0; j < 8; ++j)
                        cp[j] = e0 * v[0][j] + e1 * v[1][j] + e2 * v[2][j];
                }
                __builtin_amdgcn_sched_barrier(0);
            }
        }
        __syncthreads();

        // ============= Stage 3: attended = ctx @ out_proj_w^T + out_proj_b ===
        #pragma unroll
        for (int t = 0; t < 3; ++t) {
            v2f a[8];
            #pragma unroll
            for (int c = 0; c < 8; ++c)
                a[c] = *(const v2f*)&ctxs[(l16 * 3 + t) * 32 + 4 * c + 2 * lhalf];

            #pragma unroll
            for (int ct = 0; ct < 2; ++ct) {
                v8f acc = {};
                #pragma unroll
                for (int c = 0; c < 8; ++c)
                    acc = __builtin_amdgcn_wmma_f32_16x16x4_f32(
                            false, a[c], false, wb_out[ct][c], (short)0, acc, false, false);
                #pragma unroll
                for (int r = 0; r < 8; ++r) {
                    const int m = 8 * lhalf + r;
                    // 'attended' reuses the (now dead) qkv LDS region
                    qkvs[(m * 3 + t) * 32 + 16 * ct + l16] = acc[r] + bias_out[ct];
                }
            }
            __builtin_amdgcn_sched_barrier(0);
        }
        __syncthreads();

        // ============= Stage 4: weighted residual LN + fuse (per-sample) =====
        if (lane < 16) {
            const int s  = lane;
            const int sg = sbase + s;
            const int mk0 = masks[(size_t)sg * 3 + 0] != 0;
            const int mk1 = masks[(size_t)sg * 3 + 1] != 0;
            const int mk2 = masks[(size_t)sg * 3 + 2] != 0;
            const int mk[3] = { mk0, mk1, mk2 };
            const int count = mk0 + mk1 + mk2;

            // masked softmax over modal_weights
            float lg[3];
            #pragma unroll
            for (int i = 0; i < 3; ++i) lg[i] = mk[i] ? mweights[i] : NEGV;
            float wmax = fmaxf(lg[0], fmaxf(lg[1], lg[2]));
            float w0 = __expf(lg[0] - wmax), w1 = __expf(lg[1] - wmax), w2 = __expf(lg[2] - wmax);
            float winv = 1.0f / (w0 + w1 + w2);
            const float w[3] = { w0 * winv, w1 * winv, w2 * winv };

            float fused[32], single[32];
            #pragma unroll
            for (int d = 0; d < 32; ++d) { fused[d] = 0.f; single[d] = 0.f; }

            #pragma unroll
            for (int t = 0; t < 3; ++t) {
                const float* fr = FEAT(t) + (size_t)sg * 32;
                float x[32];
                float sum = 0.f;
                #pragma unroll
                for (int d = 0; d < 32; ++d) {
                    float att = qkvs[(s * 3 + t) * 32 + d];   // attended
                    float f   = fr[d];
                    float xx  = att * w[t] + f;               // weighted + residual
                    x[d] = xx;
                    sum += xx;
                    single[d] += mk[t] ? f : 0.f;
                }
                float mu = sum * (1.0f / 32.0f);
                float var = 0.f;
                #pragma unroll
                for (int d = 0; d < 32; ++d) { float dx = x[d] - mu; var += dx * dx; }
                var *= (1.0f / 32.0f);
                float inv = rsqrtf(var + LN_EPS);
                #pragma unroll
                for (int d = 0; d < 32; ++d) {
                    float o = (x[d] - mu) * inv * gamma[d] + beta[d];
                    fused[d] += o * w[t];
                }
                __builtin_amdgcn_sched_barrier(0);
            }

            float* op = out + (size_t)sg * 32;
            #pragma unroll
            for (int d = 0; d < 32; ++d) {
                float r = fused[d];
                if (count == 1) r = single[d];
                if (count == 0) r = 0.f;
                op[d] = r;
            }
        }
        __syncthreads();   // protect LDS reuse across group iterations
    }
}

extern "C" void kernel_launch(void* const* d_in, const int* in_sizes, int n_in,
                              void* d_out, int out_size, void* d_ws, size_t ws_size,
                              hipStream_t stream) {
    const float* metric   = (const float*)d_in[0];
    const float* logm     = (const float*)d_in[1];
    const float* trace    = (const float*)d_in[2];
    const int*   masks    = (const int*)  d_in[3];
    const float* in_w     = (const float*)d_in[4];
    const float* in_b     = (const float*)d_in[5];
    const float* out_w    = (const float*)d_in[6];
    const float* out_b    = (const float*)d_in[7];
    const float* mweights = (const float*)d_in[8];
    const float* gamma    = (const float*)d_in[9];
    const float* beta     = (const float*)d_in[10];
    float*       out      = (float*)d_out;

    const int B = in_sizes[0] / 32;                 // 262144
    const int blocks = B / SAMPLES_PER_BLOCK;       // 2048
    cmaf_kernel<<<blocks, WAVES_PER_BLOCK * 32, 0, stream>>>(
        metric, logm, trace, masks, in_w, in_b, out_w, out_b,
        mweights, gamma, beta, out);
}